// HierarchicalSoftmaxMappingModule_43928925504269
// MI455X (gfx1250) — compile-verified
//
#include <hip/hip_runtime.h>
#include <math.h>

typedef __attribute__((ext_vector_type(2))) float v2f;
typedef __attribute__((ext_vector_type(8))) float v8f;

// ---------------------------------------------------------------------------
// Kernel 1: per-row hierarchical-softmax loss.
// One 16-lane team per row; lane t handles tree level t (t < L == 15).
// Closed-form tree index removes the serial bit-scan:
//   col_t = (1<<t) - 1 + (ci >> (L - t)),  bit_t = (ci >> (L-1-t)) & 1
//   term  = softplus(bit ? s : -s)   [== -log of that level's factor]
// Covers Bpad rows (Bpad = round_up(B,64)); padding rows write 0 so the
// WMMA reduction kernel can run guard-free.
// ---------------------------------------------------------------------------
__global__ void hsm_row_loss(const float* __restrict__ scores,
                             const int* __restrict__ cls,
                             float* __restrict__ row_loss,
                             int B, int V, int L) {
    int gid = blockIdx.x * blockDim.x + threadIdx.x;
    int row = gid >> 4;        // 16 lanes per row
    int t   = gid & 15;

    float term = 0.0f;
    if (row < B && t < L) {
        int ci  = cls[row];
        int col = (1 << t) - 1 + (ci >> (L - t));
        float s = scores[(long long)row * V + col];
        int bit = (ci >> (L - 1 - t)) & 1;
        float x = bit ? s : -s;
        // stable softplus: max(x,0) + log1p(exp(-|x|))
        term = fmaxf(x, 0.0f) + log1pf(expf(-fabsf(x)));
    }

    // butterfly reduction across the 16-lane team (wave32, width 16)
    #pragma unroll
    for (int m = 8; m >= 1; m >>= 1)
        term += __shfl_xor(term, m, 16);

    if (t == 0)
        row_loss[row] = term;      // rows >= B (padding) store 0.0f
}

// ---------------------------------------------------------------------------
// Kernel 2: deterministic final reduction via V_WMMA_F32_16X16X4_F32.
// One wave, uniform control flow (EXEC all-ones as WMMA requires).
// Each iteration: one coalesced global_load_b64 per lane fills the 16x4 fp32
// A matrix with 64 values; B = ones(4x16); C accumulates D[m,n] = sum_k A[m,k].
// Placement of values inside A is irrelevant for the grand total, which is
// sum_m D[m,0]: column 0 of C lives in lane 0 (M=0..7, VGPRs 0..7) and
// lane 16 (M=8..15). Fixed order -> bit-identical across graph replays.
// ---------------------------------------------------------------------------
__global__ void hsm_reduce_wmma(const float* __restrict__ row_loss,
                                float* __restrict__ out, int B, int Bpad) {
    int lane = threadIdx.x;            // block == exactly one wave32
    v8f c = {};
    v2f bmat;
    bmat.x = 1.0f;
    bmat.y = 1.0f;

    const v2f* __restrict__ p = (const v2f*)row_loss;  // 8-byte packs
    int iters = Bpad >> 6;                             // 64 values / iter
    for (int i = 0; i < iters; ++i) {
        v2f a = p[i * 32 + lane];                      // global_load_b64
        // D = A x ones(4x16) + C   (fp32 WMMA, RNE)
        c = __builtin_amdgcn_wmma_f32_16x16x4_f32(
                /*neg_a=*/false, a, /*neg_b=*/false, bmat,
                /*c_mod=*/(short)0, c, /*reuse_a=*/false, /*reuse_b=*/false);
    }

    float partial = c[0] + c[1] + c[2] + c[3] + c[4] + c[5] + c[6] + c[7];
    float other   = __shfl_xor(partial, 16, 32);   // pair lane0 <-> lane16
    if (lane == 0)
        out[0] = (partial + other) / (float)B;
}

// ---------------------------------------------------------------------------
extern "C" void kernel_launch(void* const* d_in, const int* in_sizes, int n_in,
                              void* d_out, int out_size, void* d_ws, size_t ws_size,
                              hipStream_t stream) {
    const float* scores = (const float*)d_in[0];
    const int*   cls    = (const int*)d_in[1];

    int B = in_sizes[1];                       // 2048
    long long total = (long long)in_sizes[0];  // B * V
    int V = (int)(total / B);                  // 32768
    int L = 0;
    while ((1LL << L) < V) ++L;                // ceil(log2(V)) == 15

    int Bpad = (B + 63) & ~63;                 // multiple of 64 for WMMA loop
    float* row_loss = (float*)d_ws;            // Bpad floats of scratch

    int threads = 256;                                    // 8 waves / block
    int blocks  = (Bpad * 16 + threads - 1) / threads;    // 16 lanes per row
    hsm_row_loss<<<blocks, threads, 0, stream>>>(scores, cls, row_loss, B, V, L);

    hsm_reduce_wmma<<<1, 32, 0, stream>>>(row_loss, (float*)d_out, B, Bpad);
}